// TNLayer_30374008718006
// MI455X (gfx1250) — compile-verified
//
#include <hip/hip_runtime.h>
#include <hip/hip_bf16.h>

typedef __attribute__((ext_vector_type(8))) float v8f;
typedef __attribute__((ext_vector_type(2))) float v2f;

#define CHI 4
#define DM 16            // CHI*CHI
#define NOBS 256
#define NCOLS 255        // NVERT (param matrix columns)
#define NOFF 120         // D*(D-1)/2
#define GB 8             // batch elements per workgroup

// Packed per-layer W: fragment-ready planes.
//   float2 Wpack[l][c=64][khalf=2][n=16][p=3]   p: 0=bre, 1=bim, 2=bsum(=bre+bim)
//   bre for (c,khalf,n) = { Wre[c*4+khalf][n], Wre[c*4+khalf+2][n] }
#define WPACK_F2_PER_LAYER (64 * 2 * 16 * 3)     // 6144 float2 = 48 KB

// ---------------------------------------------------------------------------
// Kernel 1: build U[l] = expm(i*H_l) for l = 0..7 (only first 8 vertices used).
// ---------------------------------------------------------------------------
__global__ void __launch_bounds__(256) build_unitaries(const float* __restrict__ params,
                                                       float2* __restrict__ Uout) {
    __shared__ float2 A[DM][DM], P[DM][DM], T[DM][DM], Uacc[DM][DM];
    __shared__ int sS;
    const int l = blockIdx.x;
    const int t = threadIdx.x;
    const int i = t >> 4, j = t & 15;

    float hre, him;
    if (i < j) {
        int idx = i * DM - (i * (i + 1)) / 2 + (j - i - 1);
        hre = params[idx * NCOLS + l];
        him = params[(NOFF + idx) * NCOLS + l];
    } else if (i > j) {
        int idx = j * DM - (j * (j + 1)) / 2 + (i - j - 1);
        hre = params[idx * NCOLS + l];
        him = -params[(NOFF + idx) * NCOLS + l];
    } else {
        him = 0.0f;
        if (i < DM - 1) {
            hre = params[(2 * NOFF + i) * NCOLS + l];
        } else {
            float s = 0.0f;
            for (int d = 0; d < DM - 1; ++d) s += params[(2 * NOFF + d) * NCOLS + l];
            hre = -s;
        }
    }
    A[i][j].x = -him;   // (i*H).re
    A[i][j].y = hre;    // (i*H).im
    __syncthreads();

    if (t == 0) {
        float mx = 0.0f;
        for (int r = 0; r < DM; ++r) {
            float s = 0.0f;
            for (int c = 0; c < DM; ++c) s += fabsf(A[r][c].x) + fabsf(A[r][c].y);
            mx = fmaxf(mx, s);
        }
        int S = 0;
        while (mx > 0.25f && S < 30) { mx *= 0.5f; ++S; }
        sS = S;
    }
    __syncthreads();

    const float scale = ldexpf(1.0f, -sS);
    float2 p; p.x = A[i][j].x * scale; p.y = A[i][j].y * scale;
    P[i][j] = p;
    T[i][j] = p;
    float2 u0; u0.x = (i == j ? 1.0f : 0.0f) + p.x; u0.y = p.y;
    Uacc[i][j] = u0;
    __syncthreads();

    for (int n = 2; n <= 18; ++n) {
        float rr = 0.0f, ri = 0.0f;
        for (int m = 0; m < DM; ++m) {
            float2 a = T[i][m], b = P[m][j];
            rr += a.x * b.x - a.y * b.y;
            ri += a.x * b.y + a.y * b.x;
        }
        __syncthreads();
        const float inv = 1.0f / (float)n;
        T[i][j].x = rr * inv; T[i][j].y = ri * inv;
        Uacc[i][j].x += rr * inv; Uacc[i][j].y += ri * inv;
        __syncthreads();
    }
    for (int s = 0; s < sS; ++s) {
        float rr = 0.0f, ri = 0.0f;
        for (int m = 0; m < DM; ++m) {
            float2 a = Uacc[i][m], b = Uacc[m][j];
            rr += a.x * b.x - a.y * b.y;
            ri += a.x * b.y + a.y * b.x;
        }
        __syncthreads();
        T[i][j].x = rr; T[i][j].y = ri;
        __syncthreads();
        Uacc[i][j] = T[i][j];
        __syncthreads();
    }
    Uout[l * 256 + i * DM + j] = Uacc[i][j];
}

// ---------------------------------------------------------------------------
// Kernel 2: layer superoperators, emitted directly in WMMA-B-fragment layout.
// l<7 : W[n=(d*4+g)][k=(ae*16+bf)] = sum_c U[(a4+b),(c4+d)] * conj(U[(e4+f),(c4+g)])
// l==7: W[n=(d*4+c)][k]            =       U[(a4+b),(c4+d)] * conj(U[(e4+f),(c4+d)])
// ---------------------------------------------------------------------------
__global__ void __launch_bounds__(256) build_W(const float2* __restrict__ U,
                                               float* __restrict__ Wp) {
    __shared__ float2 Us[256];
    const int l = blockIdx.x;
    const int t = threadIdx.x;
    Us[t] = U[l * 256 + t];
    __syncthreads();
    for (int e = t; e < 4096; e += 256) {
        const int k = e >> 4, n = e & 15;
        const int ae = k >> 4, bf = k & 15;
        const int a = ae >> 2, ei = ae & 3, b = bf >> 2, f = bf & 3;
        float wr, wi;
        if (l < 7) {
            const int d = n >> 2, g = n & 3;
            wr = 0.0f; wi = 0.0f;
            for (int c = 0; c < 4; ++c) {
                float2 u1 = Us[(a * 4 + b) * DM + (c * 4 + d)];
                float2 u2 = Us[(ei * 4 + f) * DM + (c * 4 + g)];
                wr += u1.x * u2.x + u1.y * u2.y;
                wi += u1.y * u2.x - u1.x * u2.y;
            }
        } else {
            const int d = n >> 2, c = n & 3;
            float2 u1 = Us[(a * 4 + b) * DM + (c * 4 + d)];
            float2 u2 = Us[(ei * 4 + f) * DM + (c * 4 + d)];
            wr = u1.x * u2.x + u1.y * u2.y;
            wi = u1.y * u2.x - u1.x * u2.y;
        }
        // Scatter into fragment-ready packing: k = c*4 + khalf + 2*sel
        const int c4    = k >> 2;
        const int r4    = k & 3;
        const int khalf = r4 & 1;
        const int sel   = r4 >> 1;            // .x or .y of the v2f fragment
        float* base = Wp + (size_t)l * (WPACK_F2_PER_LAYER * 2)
                         + ((((c4 * 2 + khalf) * DM + n) * 3) * 2 + sel);
        base[0] = wr;            // plane 0: bre
        base[2] = wi;            // plane 1: bim
        base[4] = wr + wi;       // plane 2: bsum
    }
}

// ---------------------------------------------------------------------------
// Tile pass: NT row-tiles (16 rows each) through one layer's 256-K GEMM.
// Gauss 3-mult complex accumulate on v_wmma_f32_16x16x4_f32.
// ---------------------------------------------------------------------------
template <int NT>
__device__ __forceinline__ void tile_pass(float2* __restrict__ rho,
                                          const float2* __restrict__ Wl,
                                          float* __restrict__ out,
                                          int lg, int nodes, int stride, int rows,
                                          const int* rb, int khalf, int ncol,
                                          int b0, int Btot, bool final_layer) {
    const float2* gb[NT];
    int lslot[NT], rslot[NT];
#pragma unroll
    for (int u = 0; u < NT; ++u) {
        int rm = rb[u] + ncol; if (rm >= rows) rm = rows - 1;
        const int g    = rm >> lg;
        const int node = rm & (nodes - 1);
        lslot[u] = node * (stride << 1);
        rslot[u] = lslot[u] + stride;
        gb[u] = rho + g * DM * NOBS;
    }

    // Preload the 8 right-operand complex values per tile (reused by all 16 ae).
    float2 R[NT][8];
#pragma unroll
    for (int u = 0; u < NT; ++u)
#pragma unroll
        for (int q = 0; q < 4; ++q) {
            const int bf = (q << 2) + (khalf << 1);
            R[u][q * 2 + 0] = gb[u][bf * NOBS + rslot[u]];
            R[u][q * 2 + 1] = gb[u][(bf + 1) * NOBS + rslot[u]];
        }

    v8f p1[NT], p2[NT], p3[NT];
#pragma unroll
    for (int u = 0; u < NT; ++u) { v8f z = {}; p1[u] = z; p2[u] = z; p3[u] = z; }

#pragma unroll 1
    for (int ae = 0; ae < 16; ++ae) {
        float2 L[NT];
#pragma unroll
        for (int u = 0; u < NT; ++u) L[u] = gb[u][ae * NOBS + lslot[u]];
        const float2* wbase = Wl + (((ae * 8) + khalf) * DM + ncol) * 3;
#pragma unroll
        for (int q = 0; q < 4; ++q) {
            const float2* wf = wbase + q * (2 * DM * 3);   // constant 768B steps
            float2 f0 = wf[0], f1 = wf[1], f2 = wf[2];
            v2f bre;  bre.x  = f0.x; bre.y  = f0.y;
            v2f bim;  bim.x  = f1.x; bim.y  = f1.y;
            v2f bsum; bsum.x = f2.x; bsum.y = f2.y;
#pragma unroll
            for (int u = 0; u < NT; ++u) {
                float2 r0 = R[u][q * 2 + 0], r1 = R[u][q * 2 + 1];
                v2f are, aim;
                are.x = L[u].x * r0.x - L[u].y * r0.y;  aim.x = L[u].x * r0.y + L[u].y * r0.x;
                are.y = L[u].x * r1.x - L[u].y * r1.y;  aim.y = L[u].x * r1.y + L[u].y * r1.x;
                v2f asum = are + aim;
                p1[u] = __builtin_amdgcn_wmma_f32_16x16x4_f32(false, are,  false, bre,  (short)0, p1[u], false, false);
                p2[u] = __builtin_amdgcn_wmma_f32_16x16x4_f32(false, aim,  false, bim,  (short)0, p2[u], false, false);
                p3[u] = __builtin_amdgcn_wmma_f32_16x16x4_f32(false, asum, false, bsum, (short)0, p3[u], false, false);
            }
        }
    }

    // C/D layout: VGPR jj -> M=jj (lanes 0-15) / M=jj+8 (lanes 16-31), N=lane&15.
    const int Mb = khalf << 3;
#pragma unroll
    for (int u = 0; u < NT; ++u) {
        if (!final_layer) {
#pragma unroll
            for (int jj = 0; jj < 8; ++jj) {
                const int r = rb[u] + Mb + jj;
                if (r < rows) {
                    const int g2 = r >> lg;
                    const int nd = r & (nodes - 1);
                    float2 o;
                    o.x = p1[u][jj] - p2[u][jj];                     // re
                    o.y = p3[u][jj] - p1[u][jj] - p2[u][jj];         // im
                    rho[(g2 * DM + ncol) * NOBS + nd * (stride << 1)] = o;  // in-place: left slot
                }
            }
        } else {
            for (int jj = 0; jj < 8; ++jj) {
                const int r = rb[u] + Mb + jj;
                if (r < rows && (b0 + r) < Btot)
                    out[(size_t)(b0 + r) * DM + ncol] = p1[u][jj] - p2[u][jj];
            }
        }
    }
}

// ---------------------------------------------------------------------------
// Kernel 3: whole-tree contraction; tree resident in LDS, W async-staged.
// ---------------------------------------------------------------------------
__global__ void __launch_bounds__(256) tn_contract(const float* __restrict__ in_re,
                                                   const float* __restrict__ in_im,
                                                   const float* __restrict__ Wp,
                                                   float* __restrict__ out,
                                                   int Btot) {
    extern __shared__ char smem[];
    float2* rho = (float2*)smem;                                      // [GB][16][256], interleaved re/im
    float2* Wl  = (float2*)(smem + GB * DM * NOBS * sizeof(float2));  // packed planes, 48 KB

    const int tid  = threadIdx.x;
    const int wave = tid >> 5;
    const int lane = tid & 31;
    const int b0   = blockIdx.x * GB;

    // Leaves: rho[g][j=a*4+e][k] = re/im[b, a, e, k]  (k coalesced).
    for (int g = 0; g < GB; ++g) {
        const bool bok = (b0 + g) < Btot;
        const float* rb = in_re + (size_t)(b0 + g) * (DM * NOBS);
        const float* ib = in_im + (size_t)(b0 + g) * (DM * NOBS);
#pragma unroll
        for (int j = 0; j < DM; ++j) {
            float2 v;
            v.x = bok ? rb[j * NOBS + tid] : 0.0f;
            v.y = bok ? ib[j * NOBS + tid] : 0.0f;
            rho[(g * DM + j) * NOBS + tid] = v;
        }
    }

    const int khalf = lane >> 4;   // ISA 16x4 A layout: lanes 0-15 hold K=0,1; 16-31 hold K=2,3
    const int ncol  = lane & 15;
    const unsigned wlBase = (unsigned)(uintptr_t)(void*)Wl;  // LDS byte offset of W staging area

#pragma unroll
    for (int l = 0; l < 8; ++l) {
        __syncthreads();
        // Stage this layer's packed W (48 KB contiguous) via async global->LDS DMA.
        {
            const char* gsrc = (const char*)(Wp + (size_t)l * (WPACK_F2_PER_LAYER * 2));
            for (int idx = tid; idx < WPACK_F2_PER_LAYER / 2; idx += 256) {   // 3072 x 16B
                unsigned           lo = wlBase + (unsigned)idx * 16u;
                unsigned long long ga = (unsigned long long)(uintptr_t)(gsrc + (size_t)idx * 16u);
                asm volatile("global_load_async_to_lds_b128 %0, %1, off"
                             :: "v"(lo), "v"(ga) : "memory");
            }
            asm volatile("s_wait_asynccnt 0" ::: "memory");
        }
        __syncthreads();

        const int stride = 1 << l;          // input slot stride
        const int nodes  = 128 >> l;        // output nodes this layer
        const int rows   = GB * nodes;
        const int ntiles = (rows + 15) >> 4;
        const int lg     = 7 - l;           // log2(nodes)
        const bool fin   = (l == 7);

        // Paired row-tiles (M=32 blocking): amortizes B-fragment loads, 6 acc chains.
        const int pairs = ntiles >> 1;
        for (int pt = wave; pt < pairs; pt += 8) {
            int rb2[2] = { (pt * 2) << 4, (pt * 2 + 1) << 4 };
            tile_pass<2>(rho, Wl, out, lg, nodes, stride, rows, rb2, khalf, ncol, b0, Btot, fin);
        }
        // Odd tail (l==6 and l==7: single 16-row tile).
        if ((ntiles & 1) && wave == 0) {
            int rb1[1] = { (ntiles - 1) << 4 };
            tile_pass<1>(rho, Wl, out, lg, nodes, stride, rows, rb1, khalf, ncol, b0, Btot, fin);
        }
    }
}

// ---------------------------------------------------------------------------
extern "C" void kernel_launch(void* const* d_in, const int* in_sizes, int n_in,
                              void* d_out, int out_size, void* d_ws, size_t ws_size,
                              hipStream_t stream) {
    const float* in_re  = (const float*)d_in[0];
    const float* in_im  = (const float*)d_in[1];
    const float* params = (const float*)d_in[2];
    float* out = (float*)d_out;

    const int Btot = in_sizes[0] / (DM * NOBS);   // 4096

    // Workspace: U (8 x 16x16 complex = 16 KB) | packed W (8 x 48 KB = 384 KB)
    float2* U = (float2*)d_ws;
    float*  W = (float*)((char*)d_ws + 8 * 256 * sizeof(float2));

    build_unitaries<<<8, 256, 0, stream>>>(params, U);
    build_W<<<8, 256, 0, stream>>>(U, W);

    const int grid = (Btot + GB - 1) / GB;        // 512 workgroups
    const size_t lds = (size_t)GB * DM * NOBS * sizeof(float2)        // 256 KB rho
                     + (size_t)WPACK_F2_PER_LAYER * sizeof(float2);   // 48 KB W planes
    tn_contract<<<grid, 256, lds, stream>>>(in_re, in_im, W, out, Btot);
}